// GraphSAGE_23768349016495
// MI455X (gfx1250) — compile-verified
//
#include <hip/hip_runtime.h>
#include <hip/hip_bf16.h>

#define NN 500000
#define NE 16000000

typedef __attribute__((ext_vector_type(2))) float v2f;
typedef __attribute__((ext_vector_type(8))) float v8f;

#if defined(__gfx1250__) && __has_builtin(__builtin_amdgcn_wmma_f32_16x16x4_f32)
#define HAVE_WMMA_F32X4 1
#else
#define HAVE_WMMA_F32X4 0
#endif

// ---------------- degree / deg_inv ----------------

__global__ void k_degree(const int* __restrict__ dst, float* __restrict__ deg) {
  int e = blockIdx.x * blockDim.x + threadIdx.x;
  if (e < NE) atomicAdd(&deg[dst[e]], 1.0f);
}

__global__ void k_deginv(const float* __restrict__ deg, float* __restrict__ dinv) {
  int i = blockIdx.x * blockDim.x + threadIdx.x;
  if (i < NN) dinv[i] = 1.0f / fmaxf(deg[i], 1.0f);
}

// ---------------- weight packing: W[FOUT][FIN] -> Wp[32][KPAD] zero-padded ----
// Wp[n][kk] = (n<FOUT && kk<FIN) ? W[n][kk] : 0.  B-operand loads become
// unconditional + in-bounds -> no divergent exec diamonds in the hot kernel.

__global__ void k_packw(const float* __restrict__ W, float* __restrict__ Wp,
                        int FOUT, int FIN, int KPAD) {
  int t = blockIdx.x * blockDim.x + threadIdx.x;
  if (t >= 32 * KPAD) return;
  int n = t / KPAD, kk = t % KPAD;
  Wp[t] = (n < FOUT && kk < FIN) ? W[n * FIN + kk] : 0.0f;
}

__global__ void k_packb(const float* __restrict__ b, float* __restrict__ bp, int FOUT) {
  int t = threadIdx.x;
  if (t < 32) bp[t] = (t < FOUT) ? b[t] : 0.0f;
}

// ---------------- edge scatter: agg[dst] += h[src] ----------------
// STRIDE is padded row stride (multiple of 4 floats -> 16B-aligned float4 loads)

template<int F, int STRIDE>
__global__ void k_scatter(const int* __restrict__ src, const int* __restrict__ dst,
                          const float* __restrict__ h, float* __restrict__ agg) {
  int e = blockIdx.x * blockDim.x + threadIdx.x;
  if (e >= NE) return;
  int s = src[e], d = dst[e];
  const float4* hs = reinterpret_cast<const float4*>(h + (size_t)s * STRIDE);
  float* ad = agg + (size_t)d * STRIDE;
#pragma unroll
  for (int f = 0; f < F; f += 4) {
    float4 v = hs[f >> 2];
    atomicAdd(&ad[f + 0], v.x);
    if (f + 1 < F) atomicAdd(&ad[f + 1], v.y);
    if (f + 2 < F) atomicAdd(&ad[f + 2], v.z);
    if (f + 3 < F) atomicAdd(&ad[f + 3], v.w);
  }
}

// ---------------- per-node transform via V_WMMA_F32_16X16X4_F32 ----------------
// out = relu( (agg*deg_inv) @ wl.T + bias + h @ wr.T )       (AGG=true)
// out = h @ wl.T + bias                                      (AGG=false, classifier)
// Weights arrive pre-packed as Wp[32][KPAD]; bias as bp[32]. Pad columns compute
// exactly relu(0+0)=0, so no store-side masking is needed.
// One wave = one 16-node tile. Launch: grid 15625 x block 64 = 31250 tiles exact.

template<int KPAD, int FOUT, int SIN, int SOUT, bool AGG, bool RELU>
__global__ void k_transform(const float* __restrict__ h,
                            const float* __restrict__ agg,
                            const float* __restrict__ dinv,
                            const float* __restrict__ wlp,  // [32][KPAD] packed
                            const float* __restrict__ bp,   // [32] packed
                            const float* __restrict__ wrp,  // [32][KPAD] packed
                            float* __restrict__ out) {
#if HAVE_WMMA_F32X4
  const int lane = threadIdx.x & 31;
  const int wave = threadIdx.x >> 5;
  const int tile = blockIdx.x * 2 + wave;       // 31250 tiles exactly
  const int m    = lane & 15;                   // A row index
  const int hi   = lane >> 4;
  const int n    = m;                           // B/D column for this lane
  const int nodeA = tile * 16 + m;
  const float* hrow = h + (size_t)nodeA * SIN;
  const float* arow = AGG ? (agg + (size_t)nodeA * SIN) : nullptr;
  const float  di   = AGG ? dinv[nodeA] : 0.0f;

  const float* Wself = AGG ? wrp : wlp;
  const v2f* BS = reinterpret_cast<const v2f*>(Wself);  // [32][KPAD] as v2f pairs
  const v2f* BL = reinterpret_cast<const v2f*>(wlp);

  v8f c1 = {0.f,0.f,0.f,0.f,0.f,0.f,0.f,0.f};
  v8f c2 = {0.f,0.f,0.f,0.f,0.f,0.f,0.f,0.f};

#pragma unroll
  for (int k = 0; k < KPAD; k += 4) {
    const int kk = k + 2 * hi;   // A layout: vgpr0 = K(2hi), vgpr1 = K(2hi+1)
    // ---- self path: A = h rows, B = Wself.T ----
    v2f a  = *reinterpret_cast<const v2f*>(hrow + kk);          // 8B-aligned
    v2f bs = BS[(n * KPAD + kk) >> 1];
    c1 = __builtin_amdgcn_wmma_f32_16x16x4_f32(false, a, false, bs, (short)0, c1, false, false);
    if constexpr (FOUT > 16) {
      v2f bs2 = BS[((n + 16) * KPAD + kk) >> 1];
      c2 = __builtin_amdgcn_wmma_f32_16x16x4_f32(false, a, false, bs2, (short)0, c2, false, false);
    }
    // ---- aggregate path: A = agg*deg_inv rows, B = wl.T ----
    if constexpr (AGG) {
      v2f aa = *reinterpret_cast<const v2f*>(arow + kk);
      aa.x *= di; aa.y *= di;
      v2f bl = BL[(n * KPAD + kk) >> 1];
      c1 = __builtin_amdgcn_wmma_f32_16x16x4_f32(false, aa, false, bl, (short)0, c1, false, false);
      if constexpr (FOUT > 16) {
        v2f bl2 = BL[((n + 16) * KPAD + kk) >> 1];
        c2 = __builtin_amdgcn_wmma_f32_16x16x4_f32(false, aa, false, bl2, (short)0, c2, false, false);
      }
    }
  }

  const float bn  = bp[n];                       // packed: 0 beyond FOUT
  float bn2 = 0.0f;
  if constexpr (FOUT > 16) bn2 = bp[n + 16];
  constexpr int S1 = (SOUT < 16) ? SOUT : 16;

#pragma unroll
  for (int j = 0; j < 8; ++j) {                 // D layout: vgpr j -> row j + 8*hi
    const int row = j + 8 * hi;
    float* orow = out + (size_t)(tile * 16 + row) * SOUT;
    float v = c1[j] + bn;
    if (RELU) v = fmaxf(v, 0.0f);               // pad columns: relu(0+0) == 0
    if (n < S1) orow[n] = v;
    if constexpr (FOUT > 16) {
      const int n2 = n + 16;
      float v2 = c2[j] + bn2;
      if (RELU) v2 = fmaxf(v2, 0.0f);
      if (n2 < SOUT) orow[n2] = v2;
    }
  }
#else
  // Scalar fallback (same launch geometry: 1M threads cover 500K nodes)
  const int node = blockIdx.x * blockDim.x + threadIdx.x;
  if (node >= NN) return;
  const float* hrow = h + (size_t)node * SIN;
  const float  di   = AGG ? dinv[node] : 0.0f;
  const float* arow = AGG ? (agg + (size_t)node * SIN) : nullptr;
#pragma unroll
  for (int o = 0; o < SOUT; ++o) {
    float acc = bp[o];
#pragma unroll
    for (int i = 0; i < KPAD; ++i) {
      if (AGG) acc += arow[i] * di * wlp[o * KPAD + i] + hrow[i] * wrp[o * KPAD + i];
      else     acc += hrow[i] * wlp[o * KPAD + i];
    }
    if (RELU) acc = fmaxf(acc, 0.0f);
    out[(size_t)node * SOUT + o] = acc;
  }
#endif
}

// ---------------- launch ----------------

extern "C" void kernel_launch(void* const* d_in, const int* in_sizes, int n_in,
                              void* d_out, int out_size, void* d_ws, size_t ws_size,
                              hipStream_t stream) {
  (void)in_sizes; (void)n_in; (void)out_size; (void)ws_size;
  const float* x   = (const float*)d_in[0];
  const int*   ei  = (const int*)d_in[1];     // harness delivers integers as int32
  const float* w1l = (const float*)d_in[2];
  const float* b1  = (const float*)d_in[3];
  const float* w1r = (const float*)d_in[4];
  const float* w2l = (const float*)d_in[5];
  const float* b2  = (const float*)d_in[6];
  const float* w2r = (const float*)d_in[7];
  const float* w3l = (const float*)d_in[8];
  const float* b3  = (const float*)d_in[9];
  const float* w3r = (const float*)d_in[10];
  const float* wc  = (const float*)d_in[11];
  const float* bc  = (const float*)d_in[12];
  float* out = (float*)d_out;

  const int* src = ei;
  const int* dst = ei + NE;

  char* ws = (char*)d_ws;
  size_t off = 0;
  auto take = [&](size_t bytes) -> char* {
    char* p = ws + off;
    off = (off + bytes + 255) & ~(size_t)255;
    return p;
  };
  float* deg  = (float*)take((size_t)NN * 4);
  float* dinv = (float*)take((size_t)NN * 4);
  float* h1   = (float*)take((size_t)NN * 12 * 4);  // 10 real + 2 pad
  float* h2   = (float*)take((size_t)NN * 20 * 4);
  float* h3   = (float*)take((size_t)NN * 20 * 4);
  float* agg  = (float*)take((size_t)NN * 20 * 4);  // reused per layer
  // packed weights / biases
  float* p1l = (float*)take(32 * 4  * 4);
  float* p1r = (float*)take(32 * 4  * 4);
  float* p2l = (float*)take(32 * 12 * 4);
  float* p2r = (float*)take(32 * 12 * 4);
  float* p3l = (float*)take(32 * 20 * 4);
  float* p3r = (float*)take(32 * 20 * 4);
  float* pc  = (float*)take(32 * 20 * 4);
  float* pb1 = (float*)take(32 * 4);
  float* pb2 = (float*)take(32 * 4);
  float* pb3 = (float*)take(32 * 4);
  float* pbc = (float*)take(32 * 4);

  const int EB = 256, EG = (NE + EB - 1) / EB;      // 62500
  const int NB = 256, NG = (NN + NB - 1) / NB;
  const int TG = 15625, TB = 64;                    // 31250 tiles, 2 waves/block

  // pack weights (tiny)
  k_packw<<<(32*4  + 255)/256, 256, 0, stream>>>(w1l, p1l, 10, 4,  4);
  k_packw<<<(32*4  + 255)/256, 256, 0, stream>>>(w1r, p1r, 10, 4,  4);
  k_packw<<<(32*12 + 255)/256, 256, 0, stream>>>(w2l, p2l, 20, 10, 12);
  k_packw<<<(32*12 + 255)/256, 256, 0, stream>>>(w2r, p2r, 20, 10, 12);
  k_packw<<<(32*20 + 255)/256, 256, 0, stream>>>(w3l, p3l, 20, 20, 20);
  k_packw<<<(32*20 + 255)/256, 256, 0, stream>>>(w3r, p3r, 20, 20, 20);
  k_packw<<<(32*20 + 255)/256, 256, 0, stream>>>(wc,  pc,  8,  20, 20);
  k_packb<<<1, 32, 0, stream>>>(b1, pb1, 10);
  k_packb<<<1, 32, 0, stream>>>(b2, pb2, 20);
  k_packb<<<1, 32, 0, stream>>>(b3, pb3, 20);
  k_packb<<<1, 32, 0, stream>>>(bc, pbc, 8);

  hipMemsetAsync(deg, 0, (size_t)NN * 4, stream);
  k_degree<<<EG, EB, 0, stream>>>(dst, deg);
  k_deginv<<<NG, NB, 0, stream>>>(deg, dinv);

  // layer 1: 4 -> 10 (stride 12)
  hipMemsetAsync(agg, 0, (size_t)NN * 4 * 4, stream);
  k_scatter<4, 4><<<EG, EB, 0, stream>>>(src, dst, x, agg);
  k_transform<4, 12, 4, 12, true, true><<<TG, TB, 0, stream>>>(x, agg, dinv, p1l, pb1, p1r, h1);

  // layer 2: 10 -> 20 (K padded to 12)
  hipMemsetAsync(agg, 0, (size_t)NN * 12 * 4, stream);
  k_scatter<10, 12><<<EG, EB, 0, stream>>>(src, dst, h1, agg);
  k_transform<12, 20, 12, 20, true, true><<<TG, TB, 0, stream>>>(h1, agg, dinv, p2l, pb2, p2r, h2);

  // layer 3: 20 -> 20
  hipMemsetAsync(agg, 0, (size_t)NN * 20 * 4, stream);
  k_scatter<20, 20><<<EG, EB, 0, stream>>>(src, dst, h2, agg);
  k_transform<20, 20, 20, 20, true, true><<<TG, TB, 0, stream>>>(h2, agg, dinv, p3l, pb3, p3r, h3);

  // classifier: 20 -> 8, no relu, straight to d_out
  k_transform<20, 8, 20, 8, false, false><<<TG, TB, 0, stream>>>(h3, nullptr, nullptr, pc, pbc, nullptr, out);
}